// ExperimentalTensorProduct_12962211300040
// MI455X (gfx1250) — compile-verified
//
#include <hip/hip_runtime.h>
#include <math.h>

// ============================================================================
// e3nn-style tensor product + per-sample Linear, reduced to 3 GEMMs/sample:
//   out_io[128 x (2*l3+1)] = W_cat[K,128]^T * Z_cat[K, 2*l3+1]
// K = 384 (io0), 512 (io1), 512 (io2).  HBM-bound: 738 MB of single-use
// weights @ 23.3 TB/s => ~32 us floor. Compute via V_WMMA_F32_16X16X4_F32.
// B operands are pre-swizzled in LDS into per-lane (k0,k0+1) float2 slots so
// the hot loop is: clause{2x global_load_b32 NT} + ds_load_b64 + v_wmma.
// ============================================================================

typedef float v2f __attribute__((ext_vector_type(2)));
typedef float v8f __attribute__((ext_vector_type(8)));

#define XDIM 1152
#define WROW 180224
#define ODIM 1152

// Z (B-matrix) LDS: per K-group-of-4, 32 lanes x float2, WMMA-layout swizzled.
// sizes: io0 K=384 -> 96*64, io1 K=512 -> 128*64, io2 K=512 -> 128*64 floats
#define ZB0 0
#define ZB1 6144
#define ZB2 (6144 + 8192)
#define ZTOT 22528

// Orthonormal symmetric-traceless basis (real-SH order m=-2..2: xy,yz,z2,zx,x2-y2),
// l=1 index order (y,z,x). wigner_3j(1,1,2)[i][j][k] = E[k][i][j]/sqrt(5).
__constant__ float c_E[5][3][3] = {
    {{0.f, 0.f, 0.70710678f}, {0.f, 0.f, 0.f}, {0.70710678f, 0.f, 0.f}},          // xy
    {{0.f, 0.70710678f, 0.f}, {0.70710678f, 0.f, 0.f}, {0.f, 0.f, 0.f}},          // yz
    {{-0.40824829f, 0.f, 0.f}, {0.f, 0.81649658f, 0.f}, {0.f, 0.f, -0.40824829f}},// z2
    {{0.f, 0.f, 0.f}, {0.f, 0.f, 0.70710678f}, {0.f, 0.70710678f, 0.f}},          // zx
    {{-0.70710678f, 0.f, 0.f}, {0.f, 0.f, 0.f}, {0.f, 0.f, 0.70710678f}},         // x2-y2
};

// PATHS order from the reference; weight block p lives at offset p*16384.
__constant__ int c_paths0[3] = {0, 4, 9};      // (0,0,0) (1,1,0) (2,2,0)
__constant__ int c_paths1[4] = {1, 3, 6, 8};   // (0,1,1) (1,0,1) (1,2,1) (2,1,1)
__constant__ int c_paths2[4] = {2, 5, 7, 10};  // (0,2,2) (1,1,2) (2,0,2) (2,2,2)

// Swizzled B-slot address (in floats) for Z row K, column k:
// group g = K>>2 holds 32 float2 slots; lane = (K&2)*8 + k; pair = K&1.
__device__ __forceinline__ int zslot(int K, int k) {
  return ((K >> 2) * 64) + ((K & 2) * 16) + (k * 2) + (K & 1);
}

// One io's GEMM: M=128 (wave = M-tile), N in {1,3,5}, K = NK*128.
// WMMA f32 16x16x4: A = W[u=K, w=M] 16x4 tile, B pre-swizzled from LDS.
template <int NCOL, int NK>
__device__ __forceinline__ void gemm_io(const int* __restrict__ paths,
                                        const float* __restrict__ wbase,
                                        const float* __restrict__ zb,
                                        float* __restrict__ obase,
                                        int wtile, int kh, int m, int lane,
                                        int obaseoff) {
  v8f acc = {};
  for (int q = 0; q < NK; ++q) {
    // A layout (ISA 16x4 f32): lanes 0-15 K={0,1}, lanes 16-31 K={2,3}
    const float* wp = wbase + (size_t)paths[q] * 16384 + kh * 256 + wtile * 16 + m;
    const v2f* zp = (const v2f*)zb + q * 1024 + lane;  // 32 groups * 32 slots
#pragma unroll 8
    for (int ku = 0; ku < 32; ++ku) {
      v2f a, b;
      a.x = __builtin_nontemporal_load(wp);        // W[K+0/2][w0+m], single-use stream
      a.y = __builtin_nontemporal_load(wp + 128);  // W[K+1/3][w0+m]
      b = *zp;                                     // ds_load_b64, WMMA-ready
      acc = __builtin_amdgcn_wmma_f32_16x16x4_f32(false, a, false, b, (short)0,
                                                  acc, false, false);
      wp += 512;  // 4 rows of 128 weights
      zp += 32;   // next K-group
    }
  }
  if (m < NCOL) {  // D layout: vgpr r -> row M = r + 8*kh, col = lane&15
#pragma unroll
    for (int r = 0; r < 8; ++r) {
      int wrow = wtile * 16 + r + kh * 8;
      __builtin_nontemporal_store(acc[r], obase + obaseoff + wrow * NCOL + m);
    }
  }
}

__global__ __launch_bounds__(256) void etp_wmma_kernel(
    const float* __restrict__ x, const float* __restrict__ y,
    const float* __restrict__ w, float* __restrict__ out) {
  __shared__ alignas(16) float zall[ZTOT];
  __shared__ float C222s[125];  // raw wigner-like (2,2,2) tensor
  __shared__ float yS[9];
  __shared__ float s_c222scale;

  const int n = blockIdx.x;
  const int tid = threadIdx.x;

  // --- memset padded B buffers (zero columns >= N) + small setup, in parallel
  {
    float4 z4 = make_float4(0.f, 0.f, 0.f, 0.f);
    float4* z4p = (float4*)zall;
    for (int i = tid; i < ZTOT / 4; i += 256) z4p[i] = z4;
  }
  if (tid < 9) yS[tid] = y[(size_t)n * 9 + tid];
  // Build raw (2,2,2) coupling: C[i][j][k] = 0.5*(tr(Ei Ej Ek) + tr(Ej Ei Ek)).
  if (tid < 125) {
    const int k = tid % 5, j = (tid / 5) % 5, i = tid / 25;
    float t = 0.f;
    for (int a = 0; a < 3; ++a)
      for (int b = 0; b < 3; ++b) {
        float pij = 0.f, pji = 0.f;
        for (int c = 0; c < 3; ++c) {
          pij += c_E[i][a][c] * c_E[j][c][b];
          pji += c_E[j][a][c] * c_E[i][c][b];
        }
        t += 0.5f * (pij + pji) * c_E[k][b][a];
      }
    C222s[tid] = t;
  }
  __syncthreads();
  if (tid == 0) {  // Frobenius-normalize; fold sqrt(2*l3+1)=sqrt5 and 1/sqrt(512)
    float s = 0.f;
    for (int e = 0; e < 125; ++e) s += C222s[e] * C222s[e];
    s_c222scale = 2.23606798f * 0.04419417f * rsqrtf(s);
  }
  __syncthreads();

  // ---- Phase 0: Z tensors into swizzled LDS slots (thread u = 0..127) ----
  if (tid < 128) {
    const int u = tid;
    float* zb0 = zall + ZB0;
    float* zb1 = zall + ZB1;
    float* zb2 = zall + ZB2;
    const float* xr = x + (size_t)n * XDIM;
    float x0 = xr[u];
    float x1a[3], x2a[5], y1a[3], y2a[5];
#pragma unroll
    for (int i = 0; i < 3; ++i) x1a[i] = xr[128 + 3 * u + i];
#pragma unroll
    for (int i = 0; i < 5; ++i) x2a[i] = xr[512 + 5 * u + i];
    const float y0 = yS[0];
#pragma unroll
    for (int i = 0; i < 3; ++i) y1a[i] = yS[1 + i];
#pragma unroll
    for (int i = 0; i < 5; ++i) y2a[i] = yS[4 + i];

    const float inv0 = 0.051031036f;  // 1/sqrt(384)  (fan_in io0)
    const float inv1 = 0.044194174f;  // 1/sqrt(512)  (fan_in io1/io2)

    // io0 (l3=0): (0,0,0), (1,1,0), (2,2,0)
    zb0[zslot(u, 0)] = inv0 * x0 * y0;
    float d1 = x1a[0] * y1a[0] + x1a[1] * y1a[1] + x1a[2] * y1a[2];
    zb0[zslot(128 + u, 0)] = inv0 * 0.57735027f * d1;  // 1/sqrt3
    float d2 = 0.f;
#pragma unroll
    for (int i = 0; i < 5; ++i) d2 += x2a[i] * y2a[i];
    zb0[zslot(256 + u, 0)] = inv0 * 0.44721360f * d2;  // 1/sqrt5

    // io1 (l3=1): (0,1,1), (1,0,1), (1,2,1), (2,1,1)
    const float s35 = inv1 * 0.77459667f;  // sqrt(3)/sqrt(5)
#pragma unroll
    for (int k = 0; k < 3; ++k) {
      zb1[zslot(u, k)] = inv1 * x0 * y1a[k];
      zb1[zslot(128 + u, k)] = inv1 * y0 * x1a[k];
      float a121 = 0.f, a211 = 0.f;
      for (int j = 0; j < 5; ++j) {
        float t = 0.f;
        for (int i = 0; i < 3; ++i) t += c_E[j][i][k] * x1a[i];
        a121 += y2a[j] * t;
      }
      for (int i5 = 0; i5 < 5; ++i5) {
        float t = 0.f;
        for (int j = 0; j < 3; ++j) t += c_E[i5][j][k] * y1a[j];
        a211 += x2a[i5] * t;
      }
      zb1[zslot(256 + u, k)] = s35 * a121;
      zb1[zslot(384 + u, k)] = s35 * a211;
    }

    // io2 (l3=2): (0,2,2), (1,1,2), (2,0,2), (2,2,2)
    const float s222 = s_c222scale;
#pragma unroll
    for (int k = 0; k < 5; ++k) {
      zb2[zslot(u, k)] = inv1 * x0 * y2a[k];
      float t112 = 0.f;
      for (int i = 0; i < 3; ++i)
        for (int j = 0; j < 3; ++j) t112 += c_E[k][i][j] * x1a[i] * y1a[j];
      zb2[zslot(128 + u, k)] = inv1 * t112;  // sqrt5 * (E/sqrt5)
      zb2[zslot(256 + u, k)] = inv1 * y0 * x2a[k];
      float t222 = 0.f;
      for (int i = 0; i < 5; ++i)
        for (int j = 0; j < 5; ++j) t222 += C222s[i * 25 + j * 5 + k] * x2a[i] * y2a[j];
      zb2[zslot(384 + u, k)] = s222 * t222;
    }
  }
  __syncthreads();

  // ---- Phase 1: weight-streaming WMMA GEMMs (wave = M-tile) ----
  const int wave = tid >> 5;
  const int lane = tid & 31;
  const int kh = lane >> 4;  // K-half select (lanes 16-31 hold K=2,3)
  const int m = lane & 15;   // M column (A) / N column (D)
  const float* wbase = w + (size_t)n * WROW;
  float* obase = out + (size_t)n * ODIM;

  gemm_io<1, 3>(c_paths0, wbase, zall + ZB0, obase, wave, kh, m, lane, 0);
  gemm_io<3, 4>(c_paths1, wbase, zall + ZB1, obase, wave, kh, m, lane, 128);
  gemm_io<5, 4>(c_paths2, wbase, zall + ZB2, obase, wave, kh, m, lane, 512);
}

extern "C" void kernel_launch(void* const* d_in, const int* in_sizes, int n_in,
                              void* d_out, int out_size, void* d_ws, size_t ws_size,
                              hipStream_t stream) {
  (void)n_in; (void)out_size; (void)d_ws; (void)ws_size;
  const float* x = (const float*)d_in[0];
  const float* y = (const float*)d_in[1];
  const float* w = (const float*)d_in[2];
  float* out = (float*)d_out;
  const int nsamples = in_sizes[0] / XDIM;
  etp_wmma_kernel<<<nsamples, 256, 0, stream>>>(x, y, w, out);
}